// GraphQNetwork_59210419143187
// MI455X (gfx1250) — compile-verified
//
#include <hip/hip_runtime.h>
#include <hip/hip_bf16.h>
#include <math.h>

// ---------------- problem constants ----------------
#define NUM_G   64
#define NG      1568
#define DEG     32
#define NN      (NUM_G * NG)     // 100352 nodes
#define EG      (NG * DEG)       // 50176 edges per graph
#define ETOT    (NN * DEG)       // 3211264 edges
#define HC      12
#define NH      4
#define NC      3
#define FE      8
#define SLICES  8
#define ITEMS   (EG + NG)        // edges + self loops per graph
#define CHUNK   ((ITEMS + SLICES - 1) / SLICES)

typedef float v2f __attribute__((ext_vector_type(2)));
typedef float v8f __attribute__((ext_vector_type(8)));

// ---------------- helpers ----------------
__device__ __forceinline__ void atomicMaxFloatBits(int* addr, int vbits) {
    // monotone float-max via int/uint atomics on raw float bits
    if (vbits >= 0) atomicMax(addr, vbits);
    else            atomicMin((unsigned int*)addr, (unsigned int)vbits);
}

// load a 12-float node row (48B, 16B-aligned) with 3x b128
__device__ __forceinline__ void load_row12(const float* __restrict__ p, float* v) {
    const float4* q = (const float4*)p;
    float4 a = q[0], b = q[1], c = q[2];
    v[0]=a.x; v[1]=a.y; v[2]=a.z; v[3]=a.w;
    v[4]=b.x; v[5]=b.y; v[6]=b.z; v[7]=b.w;
    v[8]=c.x; v[9]=c.y; v[10]=c.z; v[11]=c.w;
}
// load an 8-float edge row (32B, 16B-aligned) with 2x b128
__device__ __forceinline__ void load_row8(const float* __restrict__ p, float* v) {
    const float4* q = (const float4*)p;
    float4 a = q[0], b = q[1];
    v[0]=a.x; v[1]=a.y; v[2]=a.z; v[3]=a.w;
    v[4]=b.x; v[5]=b.y; v[6]=b.z; v[7]=b.w;
}

// ---------------- self-loop edge_attr fill (scatter mean over dst) ----------------
__global__ void k_loop_accum(const int* __restrict__ ei, const float* __restrict__ ea,
                             float* __restrict__ lsum, float* __restrict__ cnt) {
    int e = blockIdx.x * blockDim.x + threadIdx.x;
    if (e >= ETOT) return;
    int dst = ei[ETOT + e];
    atomicAdd(&cnt[dst], 1.0f);
    float eav[FE];
    load_row8(ea + (size_t)e * FE, eav);
    float* s = lsum + (size_t)dst * FE;
#pragma unroll
    for (int f = 0; f < FE; ++f) atomicAdd(&s[f], eav[f]);
}

__global__ void k_loop_fin(float* __restrict__ lsum, const float* __restrict__ cnt) {
    int n = blockIdx.x * blockDim.x + threadIdx.x;
    if (n >= NN) return;
    float inv = 1.0f / fmaxf(cnt[n], 1.0f);
#pragma unroll
    for (int f = 0; f < FE; ++f) lsum[(size_t)n * FE + f] *= inv;
}

__global__ void k_fill_neginf(int* __restrict__ p, int count) {
    int i = blockIdx.x * blockDim.x + threadIdx.x;
    if (i < count) p[i] = 0xFF800000; // -inf bits
}

// ---------------- node GEMM: xl = h@Wl.T + bl, xr = h@Wr.T + br (fp32 WMMA) ----------
// IN is a compile-time constant so the k-loop fully unrolls: B fragments live in
// fixed registers (no cndmask selects) and all A loads issue ahead of the WMMA chain.
template <int IN>
__global__ void k_node_gemm_wmma(const float* __restrict__ h,
                                 const float* __restrict__ Wl, const float* __restrict__ bl,
                                 const float* __restrict__ Wr, const float* __restrict__ br,
                                 float* __restrict__ xl, float* __restrict__ xr) {
    constexpr int KST = IN / 4;         // 4 (IN=16) or 3 (IN=12)
    const int lane  = threadIdx.x & 31;
    const int col   = lane & 15;        // M-row for A, N-col for B/D
    const int khalf = lane >> 4;        // 0: K={0,1}, 1: K={2,3} within a k-step
    const int wave  = blockIdx.x * (blockDim.x >> 5) + (threadIdx.x >> 5);
    const int nwav  = gridDim.x * (blockDim.x >> 5);

    // preload B fragments (Wl.T / Wr.T, K x 16 with N>=12 zero-padded)
    v2f bfl[KST], bfr[KST];
#pragma unroll
    for (int ks = 0; ks < KST; ++ks) {
        int kk = ks * 4 + khalf * 2;
        float l0 = 0.f, l1 = 0.f, r0 = 0.f, r1 = 0.f;
        if (col < HC) {
            l0 = Wl[col * IN + kk]; l1 = Wl[col * IN + kk + 1];
            r0 = Wr[col * IN + kk]; r1 = Wr[col * IN + kk + 1];
        }
        bfl[ks] = (v2f){l0, l1};
        bfr[ks] = (v2f){r0, r1};
    }
    float bll = (col < HC) ? bl[col] : 0.f;
    float brr = (col < HC) ? br[col] : 0.f;

    const int ntiles = NN >> 4; // 6272
    for (int tile = wave; tile < ntiles; tile += nwav) {
        const float* arow = h + (size_t)(tile * 16 + col) * IN;
        // issue all A-fragment loads up front
        v2f af[KST];
#pragma unroll
        for (int ks = 0; ks < KST; ++ks) {
            int kk = ks * 4 + khalf * 2;
            af[ks] = (v2f){arow[kk], arow[kk + 1]};
        }
        v8f accl = {}; v8f accr = {};
#pragma unroll
        for (int ks = 0; ks < KST; ++ks) {
            accl = __builtin_amdgcn_wmma_f32_16x16x4_f32(false, af[ks], false, bfl[ks],
                                                         (short)0, accl, false, false);
            accr = __builtin_amdgcn_wmma_f32_16x16x4_f32(false, af[ks], false, bfr[ks],
                                                         (short)0, accr, false, false);
        }
        if (col < HC) {
            // one base address per lane; rows step by 48B -> immediate store offsets
            float* pl = xl + (size_t)(tile * 16 + khalf * 8) * HC + col;
            float* pr = xr + (size_t)(tile * 16 + khalf * 8) * HC + col;
#pragma unroll
            for (int r = 0; r < 8; ++r) {
                pl[r * HC] = accl[r] + bll;
                pr[r * HC] = accr[r] + brr;
            }
        }
    }
}

// ---------------- edge pass 1: per-dst-per-head segment max of logits ----------------
__global__ void k_edge_max(const int* __restrict__ ei, const float* __restrict__ ea,
                           const float* __restrict__ loopa,
                           const float* __restrict__ xl, const float* __restrict__ xr,
                           const float* __restrict__ We, const float* __restrict__ att,
                           int* __restrict__ mG) {
    const int g = blockIdx.x / SLICES;
    const int slice = blockIdx.x % SLICES;
    const int tid = threadIdx.x;
    __shared__ int   mLoc[NG * NH];
    __shared__ float Wesh[HC * FE];
    __shared__ float attsh[HC];
    if (tid < HC * FE) Wesh[tid] = We[tid];
    if (tid < HC)      attsh[tid] = att[tid];
    for (int i = tid; i < NG * NH; i += blockDim.x) mLoc[i] = 0xFF800000;
    __syncthreads();

    const int t0 = slice * CHUNK;
    const int t1 = (t0 + CHUNK < ITEMS) ? (t0 + CHUNK) : ITEMS;
    for (int t = t0 + tid; t < t1; t += blockDim.x) {
        int src, dst; const float* eap;
        if (t < EG) { int e = g * EG + t; src = ei[e]; dst = ei[ETOT + e]; eap = ea + (size_t)e * FE; }
        else        { int n = g * NG + (t - EG); src = n; dst = n; eap = loopa + (size_t)n * FE; }
        float eav[FE], xls[HC], xrd[HC];
        load_row8(eap, eav);
        load_row12(xl + (size_t)src * HC, xls);
        load_row12(xr + (size_t)dst * HC, xrd);
        const int dl = dst - g * NG;
#pragma unroll
        for (int hh = 0; hh < NH; ++hh) {
            float lg = 0.f;
#pragma unroll
            for (int c = 0; c < NC; ++c) {
                int j = hh * NC + c;
                float ee = 0.f;
#pragma unroll
                for (int f = 0; f < FE; ++f) ee += eav[f] * Wesh[j * FE + f];
                float s = xls[j] + xrd[j] + ee;
                s = (s > 0.f) ? s : 0.2f * s;
                lg += s * attsh[j];
            }
            atomicMaxFloatBits(&mLoc[dl * NH + hh], __float_as_int(lg));
        }
    }
    __syncthreads();
    int* mg = mG + (size_t)g * NG * NH;
    for (int i = tid; i < NG * NH; i += blockDim.x) {
        int v = mLoc[i];
        if (v != (int)0xFF800000) atomicMaxFloatBits(&mg[i], v);
    }
}

// ---------------- edge pass 2: accumulate num = sum e*xl[src], den = sum e ----------------
__global__ void k_edge_acc(const int* __restrict__ ei, const float* __restrict__ ea,
                           const float* __restrict__ loopa,
                           const float* __restrict__ xl, const float* __restrict__ xr,
                           const float* __restrict__ We, const float* __restrict__ att,
                           const int* __restrict__ mG,
                           float* __restrict__ numG, float* __restrict__ denG) {
    const int g = blockIdx.x / SLICES;
    const int slice = blockIdx.x % SLICES;
    const int tid = threadIdx.x;
    extern __shared__ float dynls[];
    float* numLoc = dynls;               // NG*12
    float* denLoc = dynls + NG * HC;     // NG*4
    __shared__ float Wesh[HC * FE];
    __shared__ float attsh[HC];
    if (tid < HC * FE) Wesh[tid] = We[tid];
    if (tid < HC)      attsh[tid] = att[tid];
    for (int i = tid; i < NG * (HC + NH); i += blockDim.x) dynls[i] = 0.f;
    __syncthreads();

    const int t0 = slice * CHUNK;
    const int t1 = (t0 + CHUNK < ITEMS) ? (t0 + CHUNK) : ITEMS;
    const int* mg = mG + (size_t)g * NG * NH;
    for (int t = t0 + tid; t < t1; t += blockDim.x) {
        int src, dst; const float* eap;
        if (t < EG) { int e = g * EG + t; src = ei[e]; dst = ei[ETOT + e]; eap = ea + (size_t)e * FE; }
        else        { int n = g * NG + (t - EG); src = n; dst = n; eap = loopa + (size_t)n * FE; }
        float eav[FE], xls[HC], xrd[HC];
        load_row8(eap, eav);
        load_row12(xl + (size_t)src * HC, xls);
        load_row12(xr + (size_t)dst * HC, xrd);
        const int dl = dst - g * NG;
#pragma unroll
        for (int hh = 0; hh < NH; ++hh) {
            float lg = 0.f;
#pragma unroll
            for (int c = 0; c < NC; ++c) {
                int j = hh * NC + c;
                float ee = 0.f;
#pragma unroll
                for (int f = 0; f < FE; ++f) ee += eav[f] * Wesh[j * FE + f];
                float s = xls[j] + xrd[j] + ee;
                s = (s > 0.f) ? s : 0.2f * s;
                lg += s * attsh[j];
            }
            float mh = __int_as_float(mg[dl * NH + hh]);
            float ev = expf(lg - mh);
            atomicAdd(&denLoc[dl * NH + hh], ev);                       // ds_add_f32
#pragma unroll
            for (int c = 0; c < NC; ++c) {
                int j = hh * NC + c;
                atomicAdd(&numLoc[dl * HC + j], ev * xls[j]);           // ds_add_f32
            }
        }
    }
    __syncthreads();
    float* ng = numG + (size_t)g * NG * HC;
    float* dg = denG + (size_t)g * NG * NH;
    for (int i = tid; i < NG * HC; i += blockDim.x) atomicAdd(&ng[i], numLoc[i]);
    for (int i = tid; i < NG * NH; i += blockDim.x) atomicAdd(&dg[i], denLoc[i]);
}

// ---------------- finalize conv + GraphNorm + ReLU + feature pick ----------------
__global__ void k_norm(const float* __restrict__ numG, const float* __restrict__ denG,
                       const float* __restrict__ cb,                       // conv bias (12)
                       const float* __restrict__ nw, const float* __restrict__ nb,
                       const float* __restrict__ nms,                      // norm w,b,ms (12)
                       float* __restrict__ yv, float* __restrict__ hout,
                       float* __restrict__ zbuf, const int* __restrict__ cur, int layer) {
    const int g = blockIdx.x;
    const int tid = threadIdx.x;
    __shared__ float red[256 * HC];
    __shared__ float meansh[HC], rstdsh[HC];
    const int curn = cur[g];

    float sum[HC];
#pragma unroll
    for (int j = 0; j < HC; ++j) sum[j] = 0.f;
    for (int n = tid; n < NG; n += 256) {
        const size_t node = (size_t)g * NG + n;
        const float* nu = numG + node * HC;
        const float* de = denG + node * NH;
#pragma unroll
        for (int j = 0; j < HC; ++j) {
            float y = nu[j] / de[j / NC] + cb[j];
            yv[node * HC + j] = y;
            sum[j] += y;
        }
    }
#pragma unroll
    for (int j = 0; j < HC; ++j) red[tid * HC + j] = sum[j];
    for (int off = 128; off > 0; off >>= 1) {
        __syncthreads();
        if (tid < off) {
#pragma unroll
            for (int j = 0; j < HC; ++j) red[tid * HC + j] += red[(tid + off) * HC + j];
        }
    }
    __syncthreads();
    if (tid == 0) {
#pragma unroll
        for (int j = 0; j < HC; ++j) meansh[j] = red[j] * (1.0f / NG);
    }
    __syncthreads();

    float vsum[HC];
#pragma unroll
    for (int j = 0; j < HC; ++j) vsum[j] = 0.f;
    for (int n = tid; n < NG; n += 256) {
        const size_t node = (size_t)g * NG + n;
#pragma unroll
        for (int j = 0; j < HC; ++j) {
            float o = yv[node * HC + j] - nms[j] * meansh[j];
            vsum[j] += o * o;
        }
    }
#pragma unroll
    for (int j = 0; j < HC; ++j) red[tid * HC + j] = vsum[j];
    for (int off = 128; off > 0; off >>= 1) {
        __syncthreads();
        if (tid < off) {
#pragma unroll
            for (int j = 0; j < HC; ++j) red[tid * HC + j] += red[(tid + off) * HC + j];
        }
    }
    __syncthreads();
    if (tid == 0) {
#pragma unroll
        for (int j = 0; j < HC; ++j) rstdsh[j] = rsqrtf(red[j] * (1.0f / NG) + 1e-5f);
    }
    __syncthreads();

    for (int n = tid; n < NG; n += 256) {
        const size_t node = (size_t)g * NG + n;
#pragma unroll
        for (int j = 0; j < HC; ++j) {
            float o = yv[node * HC + j] - nms[j] * meansh[j];
            float hn = fmaxf(nw[j] * o * rstdsh[j] + nb[j], 0.f);
            hout[node * HC + j] = hn;
            if (n == curn) zbuf[g * 52 + 16 + layer * HC + j] = hn;
        }
    }
}

// ---------------- dueling head ----------------
__global__ void k_head(const float* __restrict__ x, const float* __restrict__ zbuf,
                       const int* __restrict__ cur, const int* __restrict__ mask,
                       const float* __restrict__ asW, const float* __restrict__ asb,
                       const float* __restrict__ a2W, const float* __restrict__ a2b,
                       const float* __restrict__ aW,  const float* __restrict__ ab,
                       const float* __restrict__ vsW, const float* __restrict__ vsb,
                       const float* __restrict__ v2W, const float* __restrict__ v2b,
                       const float* __restrict__ vW,  const float* __restrict__ vb,
                       float* __restrict__ out) {
    const int g = blockIdx.x, t = threadIdx.x;
    __shared__ float z[52], a1[64], a2[64], v1[16], v2s[8], adv[10], valsh;
    const int gidx = g * NG + cur[g];
    if (t < 16) z[t] = x[(size_t)gidx * 16 + t];
    for (int k = 16 + t; k < 52; k += 64) z[k] = zbuf[g * 52 + k];
    __syncthreads();
    { float acc = asb[t]; for (int k = 0; k < 52; ++k) acc += asW[t * 52 + k] * z[k]; a1[t] = fmaxf(acc, 0.f); }
    if (t < 16) { float acc = vsb[t]; for (int k = 0; k < 52; ++k) acc += vsW[t * 52 + k] * z[k]; v1[t] = fmaxf(acc, 0.f); }
    __syncthreads();
    { float acc = a2b[t]; for (int k = 0; k < 64; ++k) acc += a2W[t * 64 + k] * a1[k]; a2[t] = fmaxf(acc, 0.f); }
    if (t < 8) { float acc = v2b[t]; for (int k = 0; k < 16; ++k) acc += v2W[t * 16 + k] * v1[k]; v2s[t] = fmaxf(acc, 0.f); }
    __syncthreads();
    if (t < 10) { float acc = ab[t]; for (int k = 0; k < 64; ++k) acc += aW[t * 64 + k] * a2[k]; adv[t] = acc; }
    if (t == 0) { float acc = vb[0]; for (int k = 0; k < 8; ++k) acc += vW[k] * v2s[k]; valsh = acc; }
    __syncthreads();
    if (t < 10) {
        float ma = 0.f;
        for (int i = 0; i < 10; ++i) ma += adv[i];
        ma *= 0.1f;
        float q = valsh + adv[t] - ma;
        out[g * 10 + t] = (mask[g * 10 + t] == 0) ? -1e8f : q;
    }
}

// ---------------- host ----------------
extern "C" void kernel_launch(void* const* d_in, const int* in_sizes, int n_in,
                              void* d_out, int out_size, void* d_ws, size_t ws_size,
                              hipStream_t stream) {
    const float* x    = (const float*)d_in[0];
    const int*   ei   = (const int*)d_in[1];
    const float* ea   = (const float*)d_in[2];
    const int*   cur  = (const int*)d_in[4];
    const int*   mask = (const int*)d_in[5];

    // param pointer tables (robust to sorted-key vs insertion-order flattening)
    const float *cWe[3], *cWl[3], *cWr[3], *cAtt[3], *cB[3], *cBl[3], *cBr[3];
    const float *nW[3], *nB[3], *nMs[3];
    const float *asW, *asb, *a2W, *a2b, *aW, *ab, *vsW, *vsb, *v2W, *v2b, *vW, *vb;
    #define P(i) ((const float*)d_in[i])
    if (in_sizes[6] == 96) {
        // sorted keys: conv0..2{We,Wl,Wr,att,b,bl,br}, head(a2W,a2b,aW,ab,asW,asb,v2W,v2b,vW,vb,vsW,vsb), norm0..2{b,ms,w}
        for (int i = 0; i < 3; ++i) {
            int o = 6 + 7 * i;
            cWe[i] = P(o); cWl[i] = P(o + 1); cWr[i] = P(o + 2); cAtt[i] = P(o + 3);
            cB[i] = P(o + 4); cBl[i] = P(o + 5); cBr[i] = P(o + 6);
        }
        a2W = P(27); a2b = P(28); aW = P(29); ab = P(30); asW = P(31); asb = P(32);
        v2W = P(33); v2b = P(34); vW = P(35); vb = P(36); vsW = P(37); vsb = P(38);
        for (int i = 0; i < 3; ++i) { int o = 39 + 3 * i; nB[i] = P(o); nMs[i] = P(o + 1); nW[i] = P(o + 2); }
    } else {
        // insertion order: conv0{Wl,bl,Wr,br,We,att,b}, norm0{w,b,ms}, conv1, norm1, conv2, norm2, head(vs..,v2..,v..,as..,a2..,a..)
        int o = 6;
        for (int i = 0; i < 3; ++i) {
            cWl[i] = P(o); cBl[i] = P(o + 1); cWr[i] = P(o + 2); cBr[i] = P(o + 3);
            cWe[i] = P(o + 4); cAtt[i] = P(o + 5); cB[i] = P(o + 6); o += 7;
            nW[i] = P(o); nB[i] = P(o + 1); nMs[i] = P(o + 2); o += 3;
        }
        vsW = P(o); vsb = P(o + 1); v2W = P(o + 2); v2b = P(o + 3); vW = P(o + 4); vb = P(o + 5);
        asW = P(o + 6); asb = P(o + 7); a2W = P(o + 8); a2b = P(o + 9); aW = P(o + 10); ab = P(o + 11);
    }
    #undef P

    // workspace layout
    float* w = (float*)d_ws;
    size_t o = 0;
    float* cnt  = w + o; o += NN;           // 1
    float* loop = w + o; o += (size_t)NN * 8;
    float* xl   = w + o; o += (size_t)NN * HC;
    float* xr   = w + o; o += (size_t)NN * HC;
    int*   mG   = (int*)(w + o); o += (size_t)NN * NH;
    float* num  = w + o; o += (size_t)NN * HC;
    float* den  = w + o; o += (size_t)NN * NH;
    float* yv   = w + o; o += (size_t)NN * HC;
    float* hcur = w + o; o += (size_t)NN * HC;
    float* zbuf = w + o; o += (size_t)NUM_G * 52;

    const size_t dynBytes = (size_t)NG * (HC + NH) * sizeof(float); // ~100 KB
    (void)hipFuncSetAttribute((const void*)k_edge_acc,
                              hipFuncAttributeMaxDynamicSharedMemorySize, (int)dynBytes);

    // self-loop fill (scatter mean)
    hipMemsetAsync(cnt, 0, (size_t)NN * 9 * sizeof(float), stream); // cnt + loop contiguous
    k_loop_accum<<<(ETOT + 255) / 256, 256, 0, stream>>>(ei, ea, loop, cnt);
    k_loop_fin<<<(NN + 255) / 256, 256, 0, stream>>>(loop, cnt);

    for (int L = 0; L < 3; ++L) {
        const float* hsrc = (L == 0) ? x : hcur;
        // node GEMMs on the WMMA pipe (fp32 16x16x4), one wave per 16-row tile
        if (L == 0)
            k_node_gemm_wmma<16><<<(NN / 16) / 8, 256, 0, stream>>>(hsrc, cWl[L], cBl[L],
                                                                    cWr[L], cBr[L], xl, xr);
        else
            k_node_gemm_wmma<12><<<(NN / 16) / 8, 256, 0, stream>>>(hsrc, cWl[L], cBl[L],
                                                                    cWr[L], cBr[L], xl, xr);
        k_fill_neginf<<<(NN * NH + 255) / 256, 256, 0, stream>>>(mG, NN * NH);
        hipMemsetAsync(num, 0, (size_t)NN * (HC + NH) * sizeof(float), stream); // num+den contiguous
        k_edge_max<<<NUM_G * SLICES, 256, 0, stream>>>(ei, ea, loop, xl, xr, cWe[L], cAtt[L], mG);
        k_edge_acc<<<NUM_G * SLICES, 256, dynBytes, stream>>>(ei, ea, loop, xl, xr,
                                                              cWe[L], cAtt[L], mG, num, den);
        k_norm<<<NUM_G, 256, 0, stream>>>(num, den, cB[L], nW[L], nB[L], nMs[L],
                                          yv, hcur, zbuf, cur, L);
    }

    k_head<<<NUM_G, 64, 0, stream>>>(x, zbuf, cur, mask,
                                     asW, asb, a2W, a2b, aW, ab,
                                     vsW, vsb, v2W, v2b, vW, vb,
                                     (float*)d_out);
}